// GlobalMultiHeadAttention_27066883899847
// MI455X (gfx1250) — compile-verified
//
#include <hip/hip_runtime.h>
#include <hip/hip_bf16.h>

// GlobalMultiHeadAttention, MI455X (gfx1250).
//
// Key algebraic fact: attention is over a SINGLE summary key (K axis = 1), so
// softmax == 1.0 identically and the q/k/RoPE path cancels. Exact output:
//   S  = masked_mean_t(x)            (512 x 384)
//   SV = S @ Wv + bv                 (512x384 @ 384x384, fp32 GEMM -> WMMA f32)
//   out[bn,t,:] = x[bn,t,:] + SV[bn,:]
// Memory-bound: ~300 MB traffic -> ~13 us at 23.3 TB/s. GEMM is 151 MFLOP,
// done with V_WMMA_F32_16X16X4_F32 (exact fp32, one 16x16 tile per wave32).

#define B_ 8
#define N_ 64
#define T_ 128
#define D_ 384
#define BN_ (B_ * N_)            // 512
#define ROWTILES (BN_ / 16)      // 32
#define COLTILES (D_ / 16)       // 24

typedef __attribute__((ext_vector_type(2))) float v2f;
typedef __attribute__((ext_vector_type(8))) float v8f;

// ---------------------------------------------------------------- kernel 1
// One block per (b,n): masked mean over T=128 rows. blockDim = 384 (12 waves),
// thread d owns column d; loads are coalesced along D.
__global__ void gma_summary_kernel(const float* __restrict__ x,
                                   const int* __restrict__ masks,
                                   float* __restrict__ S) {
    __shared__ float mw[T_];
    const int bn = blockIdx.x;
    const int d  = threadIdx.x;
    const float* xb = x + (size_t)bn * T_ * D_;
    const int*   mb = masks + (size_t)bn * T_;

    if (d < T_) mw[d] = (float)mb[d];
    __syncthreads();

    float msum = 0.0f;
#pragma unroll 16
    for (int t = 0; t < T_; ++t) msum += mw[t];
    const float inv = 1.0f / fmaxf(msum, 1e-6f);

    float acc = 0.0f;
#pragma unroll 4
    for (int t = 0; t < T_; ++t) acc = fmaf(mw[t], xb[(size_t)t * D_ + d], acc);

    S[(size_t)bn * D_ + d] = acc * inv;
}

// ---------------------------------------------------------------- kernel 2
// SV = S @ Wv + bv via V_WMMA_F32_16X16X4_F32.  One wave32 per 16x16 C tile,
// 768 tiles total, K-loop 384/4 = 96 WMMA steps.
//
// Fragment layouts (CDNA5 ISA 7.12.2):
//   A (16x4, MxK):  lane L holds row m=L%16; half=L/16 selects K pair:
//                   vgpr j = A[m][2*half + j]
//   B (4x16, KxN):  assumed mirror: lane L holds col n=L%16;
//                   vgpr j = B[2*half + j][n]
//   C (16x16):      vgpr r, lane L -> C[8*half + r][L%16]
__global__ void gma_gemm_wmma_kernel(const float* __restrict__ S,
                                     const float* __restrict__ Wv,
                                     const float* __restrict__ bv,
                                     float* __restrict__ SV) {
    const int lane = threadIdx.x & 31;
    const int wave = (blockIdx.x * blockDim.x + threadIdx.x) >> 5;   // 0..767
    const int mt = wave / COLTILES;          // 0..31
    const int nt = wave % COLTILES;          // 0..23
    const int m0 = mt * 16;
    const int n0 = nt * 16;
    const int half = lane >> 4;              // 0 or 1
    const int lr   = lane & 15;              // A row / B,C col within tile

    v8f acc = {0.f, 0.f, 0.f, 0.f, 0.f, 0.f, 0.f, 0.f};

#if defined(__gfx1250__) && __has_builtin(__builtin_amdgcn_wmma_f32_16x16x4_f32)
    const float* arow = S  + (size_t)(m0 + lr) * D_ + 2 * half;      // A[m][k..]
    const float* bcol = Wv + (size_t)(2 * half) * D_ + n0 + lr;      // B[k][n]
    for (int kb = 0; kb < D_; kb += 4) {
        v2f a, b;
        a.x = arow[kb];
        a.y = arow[kb + 1];
        b.x = bcol[(size_t)kb * D_];
        b.y = bcol[(size_t)kb * D_ + D_];
        // 8 args: (neg_a, A, neg_b, B, c_mod, C, reuse_a, reuse_b)
        acc = __builtin_amdgcn_wmma_f32_16x16x4_f32(
            false, a, false, b, (short)0, acc, false, false);
    }
#else
    // Scalar fallback (host pass / missing builtin): same output mapping.
    for (int r = 0; r < 8; ++r) {
        const int m = m0 + 8 * half + r;
        const int n = n0 + lr;
        float s = 0.0f;
        for (int k = 0; k < D_; ++k)
            s = fmaf(S[(size_t)m * D_ + k], Wv[(size_t)k * D_ + n], s);
        acc[r] = s;
    }
#endif

    const float bias = bv[n0 + lr];
#pragma unroll
    for (int r = 0; r < 8; ++r) {
        const int m = m0 + 8 * half + r;
        SV[(size_t)m * D_ + n0 + lr] = acc[r] + bias;
    }
}

// ---------------------------------------------------------------- kernel 3
// out = x + broadcast(SV), float4-vectorized (384 % 4 == 0 so a float4 never
// crosses a row). Fully writes d_out.
__global__ void gma_add_kernel(const float* __restrict__ x,
                               const float* __restrict__ SV,
                               float* __restrict__ out,
                               int total4) {
    const int i4 = blockIdx.x * blockDim.x + threadIdx.x;
    if (i4 >= total4) return;
    const size_t i  = (size_t)i4 * 4;
    const int    bn = (int)(i / ((size_t)T_ * D_));
    const int    d  = (int)(i % D_);

    const float4 xv = ((const float4*)x)[i4];
    const float4 sv = ((const float4*)(SV + (size_t)bn * D_))[d >> 2];
    float4 o;
    o.x = xv.x + sv.x;
    o.y = xv.y + sv.y;
    o.z = xv.z + sv.z;
    o.w = xv.w + sv.w;
    ((float4*)out)[i4] = o;
}

// ---------------------------------------------------------------- launch
extern "C" void kernel_launch(void* const* d_in, const int* in_sizes, int n_in,
                              void* d_out, int out_size, void* d_ws, size_t ws_size,
                              hipStream_t stream) {
    const float* x     = (const float*)d_in[0];   // padded_batches (8,64,128,384)
    const float* Wv    = (const float*)d_in[7];   // (384,384)
    const float* bv    = (const float*)d_in[8];   // (384,)
    const int*   masks = (const int*)d_in[9];     // (8,64,128)
    float*       out   = (float*)d_out;

    // Workspace: S (512x384 f32) then SV (512x384 f32) = 1.5 MB total.
    float* S  = (float*)d_ws;
    float* SV = S + (size_t)BN_ * D_;

    // 1) masked mean summaries
    gma_summary_kernel<<<BN_, D_, 0, stream>>>(x, masks, S);

    // 2) SV = S @ Wv + bv  (768 tiles, 8 waves/block -> 96 blocks)
    gma_gemm_wmma_kernel<<<(ROWTILES * COLTILES) / 8, 256, 0, stream>>>(S, Wv, bv, SV);

    // 3) out = x + SV broadcast
    const int total4 = (BN_ * T_ * D_) / 4;       // 6,291,456
    gma_add_kernel<<<(total4 + 255) / 256, 256, 0, stream>>>(x, SV, out, total4);
}